// EMA_63050119905703
// MI455X (gfx1250) — compile-verified
//
#include <hip/hip_runtime.h>

// EMA-scan over t, sliced at feature 0  ==  constant 10x48 matrix W applied to
// X[t, col] where col ranges over (batch, spatial).  Computed with
// V_WMMA_F32_16X16X4_F32 (full f32), 12 chained K=4 WMMAs per 16-col tile.
//
// Memory-bound: reads only x[:, :, 0, :, :] (48 MB) + writes 10 MB.

typedef __attribute__((ext_vector_type(2))) float v2f;
typedef __attribute__((ext_vector_type(8))) float v8f;

#define NB   16
#define NT   48
#define NF   10            // window dim of x (only f==0 used)
#define SP   16384         // 128*128 spatial
#define NOUT 10            // scan outputs

#define T_STRIDE     (NF * SP)        // 163840 floats between t rows
#define B_STRIDE     (NT * NF * SP)   // 7864320 floats between batches
#define OUT_B_STRIDE (NOUT * SP)      // 163840 floats per output batch

#define TILES_TOTAL   ((NB * SP) / 16)   // 16384 tiles of 16 columns
#define TILES_PER_WAVE 4
#define WAVES_PER_BLK  8
#define NBLOCKS       (TILES_TOTAL / (TILES_PER_WAVE * WAVES_PER_BLK)) // 512

__global__ __launch_bounds__(256) void ema_wmma_kernel(const float* __restrict__ x,
                                                       float* __restrict__ out) {
    const float MU  = 2.0f / 11.0f;
    const float OMM = 9.0f / 11.0f;

    const int lane = threadIdx.x & 31;
    const int wid  = (int)((blockIdx.x * blockDim.x + threadIdx.x) >> 5);
    const int k    = lane & 15;          // output row this lane owns in A
    const bool hi  = lane >= 16;
    const int koff0 = hi ? 2 : 0;        // K offsets {0,1} lo-half, {2,3} hi-half

    // ---- per-lane constants: W[k][t] = p_k * c_t + q_k * [t==47] ----------
    // o_0 = L; o_1 = OMM*L + MU*x47; o_k = OMM*o_{k-1} + MU*o_{k-2}
    float p0 = 1.f, q0 = 0.f;            // o_0 coefficients (L, x47)
    float p1 = OMM, q1 = MU;             // o_1
    float p = (k == 0) ? p0 : p1;
    float q = (k == 0) ? q0 : q1;
#pragma unroll
    for (int j = 2; j < 16; ++j) {
        float pn = OMM * p1 + MU * p0;
        float qn = OMM * q1 + MU * q0;
        p0 = p1; q0 = q1; p1 = pn; q1 = qn;
        if (k == j) { p = pn; q = qn; }
    }
    if (k >= NOUT) { p = 0.f; q = 0.f; } // pad rows 10..15 of W with zeros

    // c_t = OMM^48/48 + MU*OMM^(47-t)   (EMA over 48 steps starting at mean)
    float omm48 = 1.f;
#pragma unroll
    for (int i = 0; i < NT; ++i) omm48 *= OMM;
    const float cbase = omm48 * (1.0f / (float)NT);

    float wa[12], wb[12];                // A-fragment values per K-chunk
    {
        float pw = 1.f;                  // OMM^(47-t), t descending
#pragma unroll
        for (int t = NT - 1; t >= 0; --t) {
            float ct = cbase + MU * pw;
            float w  = p * ct + ((t == NT - 1) ? q : 0.f);
            int ci = t >> 2, off = t & 3;
            if (off == koff0)          wa[ci] = w;
            else if (off == koff0 + 1) wb[ci] = w;
            pw *= OMM;
        }
    }

    const int n = hi ? (lane - 16) : lane;   // column within tile this lane serves

    // ---- stream tiles: D(16x16) += A(16x4) x B(4x16), 12 chunks of K=4 ----
#pragma unroll 1
    for (int it = 0; it < TILES_PER_WAVE; ++it) {
        const int tile = wid * TILES_PER_WAVE + it;
        const int col0 = tile << 4;
        const int b    = col0 >> 14;         // / SP  (tiles never straddle batch)
        const int s0   = col0 & (SP - 1);
        const float* px = x + (long)b * B_STRIDE + s0 + n;   // f==0 plane

        v8f c = {0.f, 0.f, 0.f, 0.f, 0.f, 0.f, 0.f, 0.f};
#pragma unroll
        for (int ci = 0; ci < 12; ++ci) {
            const int t0 = ci * 4 + koff0;
            v2f bfrag;
            bfrag.x = px[(long)t0 * T_STRIDE];
            bfrag.y = px[(long)(t0 + 1) * T_STRIDE];
            v2f afrag;
            afrag.x = wa[ci];
            afrag.y = wb[ci];
            // 8 args: (neg_a, A, neg_b, B, c_mod, C, reuse_a, reuse_b)
            c = __builtin_amdgcn_wmma_f32_16x16x4_f32(false, afrag, false, bfrag,
                                                      (short)0, c, false, false);
        }

        // D layout: lanes 0-15 VGPR r = row r; lanes 16-31 VGPR r = row 8+r
        float* po = out + (long)b * OUT_B_STRIDE + s0 + n;
        if (!hi) {
#pragma unroll
            for (int r = 0; r < 8; ++r) po[r * SP] = c[r];
        } else {
            po[8 * SP] = c[0];
            po[9 * SP] = c[1];
        }
    }
}

extern "C" void kernel_launch(void* const* d_in, const int* in_sizes, int n_in,
                              void* d_out, int out_size, void* d_ws, size_t ws_size,
                              hipStream_t stream) {
    (void)in_sizes; (void)n_in; (void)d_ws; (void)ws_size; (void)out_size;
    const float* x = (const float*)d_in[0];
    float* out = (float*)d_out;
    ema_wmma_kernel<<<dim3(NBLOCKS), dim3(256), 0, stream>>>(x, out);
}